// RefCondMul_13039520711162
// MI455X (gfx1250) — compile-verified
//
#include <hip/hip_runtime.h>
#include <hip/hip_bf16.h>

// ---------------------------------------------------------------------------
// Expert-grouped tiny-GEMM for per-token 1x16 @ 16x16 + bias (MoE-style).
//   out[t, n] = sum_m x[t, m] * w[inds[t], m, n] + b[inds[t], n]
// Phase 1-3: bucket tokens by expert (histogram -> scan -> scatter).
// Phase 4:   per-expert GEMM on 16-token tiles using V_WMMA_F32_16X16X4_F32.
// Memory-bound (~70 MB HBM; w/b L2-resident). Tail rows handled by CLAMPING
// the gather index (garbage A rows never affect other D rows) and guarding
// only the stores. All hot-path indices are 32-bit so gathers/scatters lower
// to the saddr + 32-bit-voffset global form (no 64-bit VALU address math).
// ---------------------------------------------------------------------------

typedef float v2f __attribute__((ext_vector_type(2)));
typedef float v8f __attribute__((ext_vector_type(8)));

#define CLASSES 1024

// ---- Phase 1: per-expert token histogram -----------------------------------
__global__ void histo_kernel(const int* __restrict__ inds,
                             int* __restrict__ counts, int n) {
    int t = blockIdx.x * blockDim.x + threadIdx.x;
    if (t < n) atomicAdd(&counts[inds[t]], 1);
}

// ---- Phase 2: exclusive scan over 1024 bins (single block) -----------------
__global__ void scan_kernel(const int* __restrict__ counts,
                            int* __restrict__ offsets,
                            int* __restrict__ cursors) {
    __shared__ int buf[2][CLASSES];
    int tid = threadIdx.x;
    buf[0][tid] = counts[tid];
    __syncthreads();
    int src = 0;
    for (int d = 1; d < CLASSES; d <<= 1) {
        int dst = 1 - src;
        int v = buf[src][tid];
        if (tid >= d) v += buf[src][tid - d];
        buf[dst][tid] = v;
        __syncthreads();
        src = dst;
    }
    int excl = (tid == 0) ? 0 : buf[src][tid - 1];
    offsets[tid] = excl;
    cursors[tid] = excl;
}

// ---- Phase 3: scatter token ids into per-expert segments -------------------
__global__ void scatter_kernel(const int* __restrict__ inds,
                               int* __restrict__ cursors,
                               int* __restrict__ sorted, int n) {
    int t = blockIdx.x * blockDim.x + threadIdx.x;
    if (t < n) {
        int e = inds[t];
        int pos = atomicAdd(&cursors[e], 1);
        sorted[pos] = t;
    }
}

// ---- Phase 4: per-expert batched GEMM via WMMA -----------------------------
// One workgroup per expert; each wave owns 16-token tiles (stride = #waves).
// D(16x16) = A(16 tokens x 16 m) * B(16 m x 16 n) + bias, K-split into 4
// chunks of V_WMMA_F32_16X16X4_F32.
__global__ __launch_bounds__(256) void moe_wmma_kernel(
    const float* __restrict__ x, const float* __restrict__ w,
    const float* __restrict__ bias, const int* __restrict__ offsets,
    const int* __restrict__ counts, const int* __restrict__ sorted,
    float* __restrict__ out) {
    const int e      = blockIdx.x;
    const int lane   = threadIdx.x & 31;
    const int wave   = threadIdx.x >> 5;
    const int nwaves = blockDim.x >> 5;
    const int col    = lane & 15;   // N index (and row M for A-side gathers)
    const int half   = lane >> 4;   // lane half selects K pair / D row half
    const int mbase  = 2 * half;    // K offset within a chunk for this half

    const int start = offsets[e];
    const int cnt   = counts[e];
    if (cnt == 0) return;           // wave-uniform: EXEC stays all-ones

    // B fragments: W[e] is 16(m) x 16(n); chunk c covers K = 4c..4c+3.
    // f32 B layout (mirrors 32-bit 16x4 A): lanes 0-15 hold K=4c+{0,1} in
    // {v.x,v.y}; lanes 16-31 hold K=4c+{2,3}.
    const int wbase = e * 256;
    v2f bfrag[4];
#pragma unroll
    for (int c = 0; c < 4; ++c) {
        int m = 4 * c + mbase;
        bfrag[c].x = w[wbase + m * 16 + col];
        bfrag[c].y = w[wbase + (m + 1) * 16 + col];
    }
    const float bv = bias[e * 16 + col];

    const int ntiles = (cnt + 15) >> 4;
    for (int tile = wave; tile < ntiles; tile += nwaves) {
        const int  tbase = tile * 16;
        const bool full  = (tbase + 16 <= cnt);

        // A-side gather row M = col (duplicated across lane halves). For the
        // tail tile, clamp the index: garbage rows are computed but never
        // stored, and cannot affect other rows of D.
        int idx = tbase + col;
        if (!full && idx >= cnt) idx = cnt - 1;
        const int tok    = sorted[start + idx];
        const int tokoff = tok << 4;  // 32-bit element offset into x / out

        // C init = bias broadcast over all 16 rows (D row-major layout:
        // VGPR r <-> row r + 8*half, lane&15 <-> column n).
        v8f acc;
#pragma unroll
        for (int i = 0; i < 8; ++i) acc[i] = bv;

        // Unconditional, branch-free A loads (4x global_load_b64 per lane,
        // saddr + 32-bit voffset form).
        v2f afrag[4];
#pragma unroll
        for (int c = 0; c < 4; ++c) {
            int m      = 4 * c + mbase;
            afrag[c].x = x[tokoff + m];
            afrag[c].y = x[tokoff + m + 1];
        }

#pragma unroll
        for (int c = 0; c < 4; ++c) {
            acc = __builtin_amdgcn_wmma_f32_16x16x4_f32(
                /*neg_a=*/false, afrag[c], /*neg_b=*/false, bfrag[c],
                /*c_mod=*/(short)0, acc, /*reuse_a=*/false, /*reuse_b=*/false);
        }

        // Recover each D-row's (pre-shifted) token offset via wave32 shuffle;
        // all lanes active, source lane `row` (0..15) holds the true value.
        int doff[8];
#pragma unroll
        for (int r = 0; r < 8; ++r)
            doff[r] = __shfl(tokoff, r + 8 * half, 32) + col;

        if (full) {
            // Fast path: unconditional stores (vast majority of tiles).
#pragma unroll
            for (int r = 0; r < 8; ++r) out[doff[r]] = acc[r];
        } else {
            // Tail tile: guard stores; condition is uniform per lane-half.
#pragma unroll
            for (int r = 0; r < 8; ++r) {
                int row = r + 8 * half;
                if (tbase + row < cnt) out[doff[r]] = acc[r];
            }
        }
    }
}

// ---------------------------------------------------------------------------
extern "C" void kernel_launch(void* const* d_in, const int* in_sizes, int n_in,
                              void* d_out, int out_size, void* d_ws,
                              size_t ws_size, hipStream_t stream) {
    const float* x    = (const float*)d_in[0];
    const int*   inds = (const int*)d_in[1];
    const float* w    = (const float*)d_in[2];
    const float* b    = (const float*)d_in[3];
    float*       out  = (float*)d_out;
    const int    n    = in_sizes[1];  // number of tokens

    char* ws      = (char*)d_ws;
    int*  counts  = (int*)(ws);            // 1024 ints
    int*  offsets = (int*)(ws + 4096);     // 1024 ints
    int*  cursors = (int*)(ws + 8192);     // 1024 ints
    int*  sorted  = (int*)(ws + 12288);    // n ints (~2 MB)

    hipMemsetAsync(counts, 0, CLASSES * sizeof(int), stream);
    histo_kernel<<<(n + 255) / 256, 256, 0, stream>>>(inds, counts, n);
    scan_kernel<<<1, CLASSES, 0, stream>>>(counts, offsets, cursors);
    scatter_kernel<<<(n + 255) / 256, 256, 0, stream>>>(inds, cursors, sorted, n);
    moe_wmma_kernel<<<CLASSES, 256, 0, stream>>>(x, w, b, offsets, counts,
                                                 sorted, out);
}